// MLKVDecoderLayer_10969346474369
// MI455X (gfx1250) — compile-verified
//
#include <hip/hip_runtime.h>
#include <hip/hip_bf16.h>
#include <math.h>

typedef __attribute__((ext_vector_type(2))) float v2f;
typedef __attribute__((ext_vector_type(8))) float v8f;

#define HIDDEN 2048
#define NHEAD 16
#define HKV 2
#define HEADD 128
#define RD 32
#define MAXSEQ 8192
#define DFF 8192
#define EPS 1e-6f
#define SCALE 0.08838834764831845f  /* 128^-0.5 */
#define GW 8                        /* waves per GEMV workgroup */

// ---------------------------------------------------------------------------
// RMSNorm over n elements, single workgroup of 256 threads.
__global__ void k_rmsnorm(const float* __restrict__ x, const float* __restrict__ w,
                          float* __restrict__ y, int n) {
  __shared__ float red[256];
  int tid = threadIdx.x;
  float s = 0.f;
  for (int i = tid; i < n; i += 256) { float v = x[i]; s += v * v; }
  red[tid] = s; __syncthreads();
  for (int o = 128; o > 0; o >>= 1) {
    if (tid < o) red[tid] += red[tid + o];
    __syncthreads();
  }
  float rs = rsqrtf(red[0] / (float)n + EPS);
  for (int i = tid; i < n; i += 256) y[i] = x[i] * rs * w[i];
}

// ---------------------------------------------------------------------------
// GEMV via V_WMMA_F32_16X16X4_F32.
// Each workgroup: 32 output columns (two 16-col WMMA tiles -> full 128B rows),
// blockIdx.y splits K into `ksplit` slices, 8 waves split the slice again.
// Partials go to part[ky*N + n]; k_gemv_reduce sums them (fixed order).
__global__ void k_gemv_wmma(const float* __restrict__ W, const float* __restrict__ x,
                            float* __restrict__ partials, int N, int K, int ksplit) {
  __shared__ float ldsX[2048];        // K/ksplit <= 2048 for all launches
  __shared__ float part[GW * 32];
  int tid = threadIdx.x;
  int ky = blockIdx.y;
  int KB = K / ksplit;                // block K-slice
  int kbase = ky * KB;
  for (int i = tid; i < KB; i += 256) ldsX[i] = x[kbase + i];
  __syncthreads();

  int wave = tid >> 5, lane = tid & 31;
  int l16 = lane & 15;
  int dof = (lane >= 16) ? 2 : 0;     // hi half-wave handles K rows +2,+3
  int n0 = blockIdx.x * 32;
  int KWV = KB / GW;                  // per-wave K range (multiple of 4)
  int kb = wave * KWV;

  const float* colA = W + (size_t)n0 + l16;        // cols n0..n0+15
  const float* colB = colA + 16;                   // cols n0+16..n0+31

  v8f accA = {};
  v8f accB = {};
#pragma unroll 4
  for (int k = kb; k < kb + KWV; k += 4) {
    int kr = k + dof;                  // row offset within slice
    size_t ra = (size_t)(kbase + kr) * N;
    size_t rb = (size_t)(kbase + kr + 1) * N;
    float b0a = colA[ra], b1a = colA[rb];
    float b0b = colB[ra], b1b = colB[rb];
    float a0 = 0.f, a1 = 0.f;
    if (l16 == 0) { a0 = ldsX[kr]; a1 = ldsX[kr + 1]; }  // lanes 0 & 16: A row 0
    v2f A = {a0, a1};
    v2f BA = {b0a, b1a};
    v2f BB = {b0b, b1b};
    accA = __builtin_amdgcn_wmma_f32_16x16x4_f32(false, A, false, BA,
                                                 (short)0, accA, false, false);
    accB = __builtin_amdgcn_wmma_f32_16x16x4_f32(false, A, false, BB,
                                                 (short)0, accB, false, false);
  }
  // D row 0 (M=0) lives in acc[0] of lanes 0..15, N = lane.
  if (lane < 16) {
    part[wave * 32 + lane]      = accA[0];
    part[wave * 32 + 16 + lane] = accB[0];
  }
  __syncthreads();
  if (tid < 32) {
    float s = 0.f;
#pragma unroll
    for (int w = 0; w < GW; ++w) s += part[w * 32 + tid];
    partials[(size_t)ky * N + n0 + tid] = s;
  }
}

// Deterministic fixed-order reduction of K-split partials (+ optional residual).
__global__ void k_gemv_reduce(const float* __restrict__ partials,
                              const float* __restrict__ res,
                              float* __restrict__ y, int N, int ksplit) {
  int i = blockIdx.x * 256 + threadIdx.x;
  if (i < N) {
    float s = 0.f;
    for (int j = 0; j < ksplit; ++j) s += partials[(size_t)j * N + i];
    if (res) s += res[i];
    y[i] = s;
  }
}

// ---------------------------------------------------------------------------
// Per-head RMSNorm + RoPE for q (16 heads from qg[:, :128]) and k (2 heads).
__global__ void k_qknorm_rope(const float* __restrict__ qg, const float* __restrict__ kraw,
                              const float* __restrict__ qw, const float* __restrict__ kw,
                              const float* __restrict__ cosv, const float* __restrict__ sinv,
                              float* __restrict__ qn, float* __restrict__ knew) {
  int b = blockIdx.x;     // 0..15 = q heads, 16..17 = kv heads
  int d = threadIdx.x;    // 128 threads
  const float* src; const float* w; float* dst;
  if (b < NHEAD) { src = qg + b * 256;          w = qw; dst = qn + b * HEADD; }
  else           { src = kraw + (b - 16) * 128; w = kw; dst = knew + (b - 16) * HEADD; }

  __shared__ float red[128];
  __shared__ float xn[128];
  float v = src[d];
  red[d] = v * v; __syncthreads();
  for (int o = 64; o > 0; o >>= 1) {
    if (d < o) red[d] += red[d + o];
    __syncthreads();
  }
  float rs = rsqrtf(red[0] / (float)HEADD + EPS);
  float nv = v * rs * w[d];
  xn[d] = nv; __syncthreads();
  float out;
  if (d < 16)       out = nv * cosv[d] - xn[d + 16] * sinv[d];
  else if (d < RD)  out = nv * cosv[d] + xn[d - 16] * sinv[d];
  else              out = nv;
  dst[d] = out;
}

// ---------------------------------------------------------------------------
// Attention scores via WMMA.  A rows = 8 query heads of one KV group,
// B cols = 16 cached K rows.  grid = (64 s-tile groups, HKV), 8 waves/block.
// Two interleaved accumulator chains break the WMMA RAW dependency.
__global__ void k_scores(const float* __restrict__ qn, const float* __restrict__ kv,
                         const float* __restrict__ knew, const float* __restrict__ mask,
                         const int* __restrict__ posp, float* __restrict__ scores) {
  int kvh = blockIdx.y;
  int wave = threadIdx.x >> 5, lane = threadIdx.x & 31;
  int tile = blockIdx.x * 8 + wave;   // exactly 512 tiles -> s in [0, 8192)
  int s0 = tile * 16;
  int pos = *posp;

  __shared__ float lq[8 * 132];       // padded rows to dodge bank conflicts
  for (int i = threadIdx.x; i < 8 * 128; i += 256) {
    int h = i >> 7, d = i & 127;
    lq[h * 132 + d] = qn[(kvh * 8 + h) * HEADD + d];
  }
  __syncthreads();

  int l16 = lane & 15;
  int dof = (lane >= 16) ? 2 : 0;
  int s = s0 + l16;
  const float* krow = (s == pos) ? (knew + kvh * HEADD)
                                 : (kv + ((size_t)kvh * MAXSEQ + s) * HEADD);
  v8f accA = {};
  v8f accB = {};
  for (int d0 = 0; d0 < HEADD; d0 += 8) {
    int da = d0 + dof;
    int db = d0 + 4 + dof;
    float b0a = krow[da],   b1a = krow[da + 1];
    float b0b = krow[db],   b1b = krow[db + 1];
    float a0a = 0.f, a1a = 0.f, a0b = 0.f, a1b = 0.f;
    if (l16 < 8) {                      // A rows 0..7 = heads; rows 8..15 zero
      a0a = lq[l16 * 132 + da]; a1a = lq[l16 * 132 + da + 1];
      a0b = lq[l16 * 132 + db]; a1b = lq[l16 * 132 + db + 1];
    }
    v2f AA = {a0a, a1a}; v2f BA = {b0a, b1a};
    v2f AB = {a0b, a1b}; v2f BB = {b0b, b1b};
    accA = __builtin_amdgcn_wmma_f32_16x16x4_f32(false, AA, false, BA,
                                                 (short)0, accA, false, false);
    accB = __builtin_amdgcn_wmma_f32_16x16x4_f32(false, AB, false, BB,
                                                 (short)0, accB, false, false);
  }
  if (lane < 16) {
    float m = mask[s];
#pragma unroll
    for (int h = 0; h < 8; ++h)
      scores[(size_t)(kvh * 8 + h) * MAXSEQ + s] = (accA[h] + accB[h]) * SCALE + m;
  }
}

// ---------------------------------------------------------------------------
// Softmax over a full 8192-row of scores, one workgroup per head.
__global__ void k_softmax(const float* __restrict__ sc, float* __restrict__ p) {
  int h = blockIdx.x, tid = threadIdx.x;
  __shared__ float red[256];
  const float* row = sc + (size_t)h * MAXSEQ;
  float* prow = p + (size_t)h * MAXSEQ;
  float mx = -1e30f;
  for (int i = tid; i < MAXSEQ; i += 256) mx = fmaxf(mx, row[i]);
  red[tid] = mx; __syncthreads();
  for (int o = 128; o > 0; o >>= 1) { if (tid < o) red[tid] = fmaxf(red[tid], red[tid + o]); __syncthreads(); }
  mx = red[0]; __syncthreads();
  float sum = 0.f;
  for (int i = tid; i < MAXSEQ; i += 256) { float e = expf(row[i] - mx); prow[i] = e; sum += e; }
  red[tid] = sum; __syncthreads();
  for (int o = 128; o > 0; o >>= 1) { if (tid < o) red[tid] += red[tid + o]; __syncthreads(); }
  float inv = 1.f / red[0];
  for (int i = tid; i < MAXSEQ; i += 256) prow[i] *= inv;
}

// ---------------------------------------------------------------------------
// attn_out[h][d] = sum_s p[h][s]*v[s][d], then multiply by sigmoid(gate[h][d]).
// grid = HKV, block = 128 threads (one per d); s==pos uses fresh v.
__global__ void k_attnv_gate(const float* __restrict__ p, const float* __restrict__ kv,
                             const float* __restrict__ vnew, const float* __restrict__ qg,
                             const int* __restrict__ posp, float* __restrict__ attng) {
  int kvh = blockIdx.x;
  int d = threadIdx.x;
  int pos = *posp;
  __shared__ float pl[8 * 512];
  const float* vbase = kv + ((size_t)(HKV + kvh) * MAXSEQ) * HEADD;
  float acc[8];
#pragma unroll
  for (int h = 0; h < 8; ++h) acc[h] = 0.f;

  for (int c0 = 0; c0 < MAXSEQ; c0 += 512) {
    __syncthreads();
    for (int i = threadIdx.x; i < 8 * 512; i += 128) {
      int h = i >> 9, s = i & 511;
      pl[h * 512 + s] = p[(size_t)(kvh * 8 + h) * MAXSEQ + c0 + s];
    }
    __syncthreads();
    for (int s = 0; s < 512; ++s) {
      int sg = c0 + s;
      const float* vrow = (sg == pos) ? (vnew + kvh * HEADD)
                                      : (vbase + (size_t)sg * HEADD);
      float vv = vrow[d];
#pragma unroll
      for (int h = 0; h < 8; ++h) acc[h] += pl[h * 512 + s] * vv;
    }
  }
#pragma unroll
  for (int h = 0; h < 8; ++h) {
    int gh = kvh * 8 + h;
    float g = qg[gh * 256 + 128 + d];            // gate half of qg
    attng[gh * HEADD + d] = acc[h] / (1.f + expf(-g));
  }
}

// ---------------------------------------------------------------------------
__global__ void k_silu_mul(const float* __restrict__ g, const float* __restrict__ u,
                           float* __restrict__ gu, int n) {
  int i = blockIdx.x * 256 + threadIdx.x;
  if (i < n) { float gv = g[i]; gu[i] = gv / (1.f + expf(-gv)) * u[i]; }
}

// ---------------------------------------------------------------------------
extern "C" void kernel_launch(void* const* d_in, const int* in_sizes, int n_in,
                              void* d_out, int out_size, void* d_ws, size_t ws_size,
                              hipStream_t stream) {
  const float* hidden = (const float*)d_in[0];
  const float* cosv   = (const float*)d_in[1];
  const float* sinv   = (const float*)d_in[2];
  const float* mask   = (const float*)d_in[3];
  const float* kvc    = (const float*)d_in[4];
  const float* Wq     = (const float*)d_in[5];
  const float* Wk     = (const float*)d_in[6];
  const float* Wv     = (const float*)d_in[7];
  const float* Wo     = (const float*)d_in[8];
  const float* qnw    = (const float*)d_in[9];
  const float* knw    = (const float*)d_in[10];
  const float* innw   = (const float*)d_in[11];
  const float* postnw = (const float*)d_in[12];
  const float* Wgate  = (const float*)d_in[13];
  const float* Wup    = (const float*)d_in[14];
  const float* Wdown  = (const float*)d_in[15];
  const int*   pos    = (const int*)d_in[16];

  float* ws = (float*)d_ws;
  float* h        = ws;                    // 2048
  float* qg       = h + 2048;              // 4096 (q|gate per head)
  float* kraw     = qg + 4096;             // 256
  float* vnew     = kraw + 256;            // 256
  float* qn       = vnew + 256;            // 2048
  float* knew     = qn + 2048;             // 256
  float* scores   = knew + 256;            // 16*8192
  float* p        = scores + 16 * MAXSEQ;  // 16*8192
  float* attng    = p + 16 * MAXSEQ;       // 2048
  float* hidden2  = attng + 2048;          // 2048
  float* hm       = hidden2 + 2048;        // 2048
  float* gvec     = hm + 2048;             // 8192
  float* uvec     = gvec + DFF;            // 8192
  float* gu       = uvec + DFF;            // 8192
  float* partials = gu + DFF;              // 16384 (max ksplit*N)

  // GEMV helper: 32-col tiles, K split across blocks, then fixed-order reduce.
  auto gemv = [&](const float* W, const float* x, const float* res, float* y,
                  int N, int K, int ksplit) {
    k_gemv_wmma<<<dim3(N / 32, ksplit), 256, 0, stream>>>(W, x, partials, N, K, ksplit);
    k_gemv_reduce<<<(N + 255) / 256, 256, 0, stream>>>(partials, res, y, N, ksplit);
  };

  // 1. input RMSNorm
  k_rmsnorm<<<1, 256, 0, stream>>>(hidden, innw, h, HIDDEN);
  // 2. QKV projections (bandwidth-bound WMMA GEMVs)
  gemv(Wq, h, nullptr, qg, 4096, HIDDEN, 4);      // 512 blocks
  gemv(Wk, h, nullptr, kraw, 256, HIDDEN, 4);
  gemv(Wv, h, nullptr, vnew, 256, HIDDEN, 4);
  // 3. per-head q/k RMSNorm + RoPE
  k_qknorm_rope<<<NHEAD + HKV, 128, 0, stream>>>(qg, kraw, qnw, knw, cosv, sinv, qn, knew);
  // 4. attention scores (WMMA), full masked sequence for static launch shape
  k_scores<<<dim3(64, HKV), 256, 0, stream>>>(qn, kvc, knew, mask, pos, scores);
  // 5. softmax per head
  k_softmax<<<NHEAD, 256, 0, stream>>>(scores, p);
  // 6. weighted V + sigmoid gating
  k_attnv_gate<<<HKV, 128, 0, stream>>>(p, kvc, vnew, qg, pos, attng);
  // 7. output projection + residual
  gemv(Wo, attng, hidden, hidden2, HIDDEN, HIDDEN, 8);   // 512 blocks
  // 8. post RMSNorm
  k_rmsnorm<<<1, 256, 0, stream>>>(hidden2, postnw, hm, HIDDEN);
  // 9. MLP
  gemv(Wgate, hm, nullptr, gvec, DFF, HIDDEN, 2);        // 512 blocks
  gemv(Wup,   hm, nullptr, uvec, DFF, HIDDEN, 2);        // 512 blocks
  k_silu_mul<<<DFF / 256, 256, 0, stream>>>(gvec, uvec, gu, DFF);
  gemv(Wdown, gu, hidden2, (float*)d_out, HIDDEN, DFF, 8); // 512 blocks
}